// SRP_map_39032662786504
// MI455X (gfx1250) — compile-verified
//
#include <hip/hip_runtime.h>
#include <hip/hip_bf16.h>
#include <stdint.h>

// SRP-PHAT steered response map.
// x:    [B*T, 144, 512] f32   (frame blocks of 288 KB)
// tau0: [144, 2048] i32        (lag index per pair x direction)
// out:  [B*T, 2048] f32        (normalized maps)

#define NPAIR   144
#define KLEN    512
#define GPTS    2048
#define THREADS 512
#define NWAVES  (THREADS / 32)
#define EPSF    1e-12f

typedef int v4i __attribute__((ext_vector_type(4)));

#if defined(__HIP_DEVICE_COMPILE__) && \
    __has_builtin(__builtin_amdgcn_global_load_async_to_lds_b128) && \
    __has_builtin(__builtin_amdgcn_s_wait_asynccnt)
#define USE_ASYNC_LDS 1
typedef __attribute__((address_space(1))) v4i g_v4i_t;   // global (AS1) b128
typedef __attribute__((address_space(3))) v4i l_v4i_t;   // LDS (AS3) b128
#else
#define USE_ASYNC_LDS 0
#endif

constexpr int VEC_TOTAL = NPAIR * KLEN / 4;         // 18432 b128 moves
constexpr int ITERS     = VEC_TOTAL / THREADS;      // 36 per thread
constexpr int CHUNKS    = 4;
constexpr int ITERS_PC  = ITERS / CHUNKS;           // 9 loads/thread/chunk
constexpr int PAIRS_PC  = NPAIR / CHUNKS;           // 36 mic pairs/chunk

// One overlap chunk: wait until this wave's outstanding async loads drop to
// WAITCNT (its first ITERS-WAITCNT loads have landed, in issue order), then a
// workgroup barrier extends that guarantee to all 16 waves, then gather the
// PAIRS_PC mic-pair rows of this chunk from LDS while later chunks stream in.
template <int WAITCNT>
__device__ __forceinline__ void chunk_gather(
    const float* __restrict__ xs, const int* __restrict__ tau0,
    int p0, int tid, float& acc0, float& acc1, float& acc2, float& acc3)
{
#if USE_ASYNC_LDS
    __builtin_amdgcn_s_wait_asynccnt(WAITCNT);
#endif
    __syncthreads();
    #pragma unroll 4
    for (int p = p0; p < p0 + PAIRS_PC; ++p) {
        const int4 iv = ((const int4*)(tau0 + p * GPTS))[tid];  // coalesced, L2-resident
        const float* __restrict__ xp = xs + p * KLEN;
        acc0 += xp[iv.x];
        acc1 += xp[iv.y];
        acc2 += xp[iv.z];
        acc3 += xp[iv.w];
    }
}

__global__ __launch_bounds__(THREADS) void srp_map_kernel(
    const float* __restrict__ x,
    const int*   __restrict__ tau0,
    float*       __restrict__ out)
{
    // 288 KB frame tile — fits the CDNA5 320 KB WGP LDS.
    __shared__ float xs[NPAIR * KLEN];
    __shared__ float red_sum[NWAVES];
    __shared__ float red_max[NWAVES];
    __shared__ float s_stats[2];   // [0]=mean, [1]=max-mean+eps

    const int tid = threadIdx.x;
    const float* __restrict__ xblk = x + (size_t)blockIdx.x * (NPAIR * KLEN);

    // ---- Stage 1: issue ALL async b128 copies global -> LDS up front ----
#if USE_ASYNC_LDS
    #pragma unroll 4
    for (int it = 0; it < ITERS; ++it) {
        const int v = it * THREADS + tid;
        __builtin_amdgcn_global_load_async_to_lds_b128(
            (g_v4i_t*)(xblk + (size_t)v * 4),
            (l_v4i_t*)(xs + v * 4),
            /*offset=*/0, /*cpol=*/0);
    }
#else
    #pragma unroll 4
    for (int it = 0; it < ITERS; ++it) {
        const int v = it * THREADS + tid;
        *(float4*)(xs + v * 4) = *(const float4*)(xblk + (size_t)v * 4);
    }
#endif

    // ---- Stage 2: chunked gather-accumulate, overlapped with the DMA ----
    float acc0 = 0.f, acc1 = 0.f, acc2 = 0.f, acc3 = 0.f;
    chunk_gather<3 * ITERS_PC>(xs, tau0, 0 * PAIRS_PC, tid, acc0, acc1, acc2, acc3);
    chunk_gather<2 * ITERS_PC>(xs, tau0, 1 * PAIRS_PC, tid, acc0, acc1, acc2, acc3);
    chunk_gather<1 * ITERS_PC>(xs, tau0, 2 * PAIRS_PC, tid, acc0, acc1, acc2, acc3);
    chunk_gather<0>           (xs, tau0, 3 * PAIRS_PC, tid, acc0, acc1, acc2, acc3);

    // ---- Stage 3: map statistics (sum for mean, max) ----
    float lsum = (acc0 + acc1) + (acc2 + acc3);
    float lmax = fmaxf(fmaxf(acc0, acc1), fmaxf(acc2, acc3));
    // wave32 butterfly reduction
    #pragma unroll
    for (int o = 16; o > 0; o >>= 1) {
        lsum += __shfl_xor(lsum, o, 32);
        lmax  = fmaxf(lmax, __shfl_xor(lmax, o, 32));
    }
    const int wave = tid >> 5;
    if ((tid & 31) == 0) { red_sum[wave] = lsum; red_max[wave] = lmax; }
    __syncthreads();
    if (tid == 0) {
        float s = red_sum[0];
        float m = red_max[0];
        #pragma unroll
        for (int w = 1; w < NWAVES; ++w) {
            s += red_sum[w];
            m  = fmaxf(m, red_max[w]);
        }
        const float mean = s * (1.0f / (float)GPTS);
        s_stats[0] = mean;
        s_stats[1] = m - mean + EPSF;   // == max(v - mean + eps)
    }
    __syncthreads();
    const float mean  = s_stats[0];
    const float denom = s_stats[1];

    // ---- Stage 4: normalize + write (coalesced b128 stores) ----
    float4 o4;
    o4.x = (acc0 - mean + EPSF) / denom;
    o4.y = (acc1 - mean + EPSF) / denom;
    o4.z = (acc2 - mean + EPSF) / denom;
    o4.w = (acc3 - mean + EPSF) / denom;
    ((float4*)(out + (size_t)blockIdx.x * GPTS))[tid] = o4;
}

extern "C" void kernel_launch(void* const* d_in, const int* in_sizes, int n_in,
                              void* d_out, int out_size, void* d_ws, size_t ws_size,
                              hipStream_t stream) {
    const float* x    = (const float*)d_in[0];
    const int*   tau0 = (const int*)d_in[1];
    float*       out  = (float*)d_out;
    const int frames  = in_sizes[0] / (NPAIR * KLEN);   // B*T = 256
    srp_map_kernel<<<frames, THREADS, 0, stream>>>(x, tau0, out);
}